// GaussianMerge_34591666602829
// MI455X (gfx1250) — compile-verified
//
#include <hip/hip_runtime.h>
#include <cstdint>

// ---------------------------------------------------------------------------
// GaussianMerge: per-element 3x3 Kalman-style merge, B = 4M.
// Memory-bound (AI ~1.4 flop/byte, 384MB in + 192MB out streamed once).
// CDNA5 path: async global->LDS / LDS->global b128 copies (ASYNCcnt),
// non-temporal TH hints since working set >> 192MB L2.
// ---------------------------------------------------------------------------

#define TILE 256

__device__ __forceinline__ uint32_t lds_off32(const void* p) {
    // flat pointer to LDS = {shared_aperture_hi32, lds_offset} -> low 32 bits
    return (uint32_t)(uintptr_t)p;
}

__device__ __forceinline__ void async_ld_b128(uint32_t lds_addr, uint32_t voff, uint64_t sbase) {
    asm volatile("global_load_async_to_lds_b128 %0, %1, %2 th:TH_LOAD_NT"
                 :: "v"(lds_addr), "v"(voff), "s"(sbase) : "memory");
}

__device__ __forceinline__ void async_st_b128(uint32_t voff, uint32_t lds_addr, uint64_t sbase) {
    asm volatile("global_store_async_from_lds_b128 %0, %1, %2 th:TH_STORE_NT"
                 :: "v"(voff), "v"(lds_addr), "s"(sbase) : "memory");
}

__device__ __forceinline__ void wait_async0() {
    asm volatile("s_wait_asynccnt 0" ::: "memory");
}

__device__ __forceinline__ void merge_one(const float a[9], const float b[9],
                                          const float l1[3], const float l2[3],
                                          float locn[3], float covn[9]) {
    const float EPS = 1e-6f;
    float S[9];
#pragma unroll
    for (int k = 0; k < 9; ++k) S[k] = a[k] + b[k];
    S[0] += EPS; S[4] += EPS; S[8] += EPS;

    // adjugate of S
    float A00 = S[4]*S[8] - S[5]*S[7];
    float A01 = S[2]*S[7] - S[1]*S[8];
    float A02 = S[1]*S[5] - S[2]*S[4];
    float A10 = S[5]*S[6] - S[3]*S[8];
    float A11 = S[0]*S[8] - S[2]*S[6];
    float A12 = S[2]*S[3] - S[0]*S[5];
    float A20 = S[3]*S[7] - S[4]*S[6];
    float A21 = S[1]*S[6] - S[0]*S[7];
    float A22 = S[0]*S[4] - S[1]*S[3];
    float det = S[0]*A00 + S[1]*A10 + S[2]*A20;
    float r   = 1.0f / det;

    float inv[9] = { A00*r, A01*r, A02*r,
                     A10*r, A11*r, A12*r,
                     A20*r, A21*r, A22*r };

    // K = cov1 @ inv(S)
    float K[9];
#pragma unroll
    for (int i = 0; i < 3; ++i)
#pragma unroll
        for (int j = 0; j < 3; ++j)
            K[i*3+j] = a[i*3+0]*inv[0*3+j] + a[i*3+1]*inv[1*3+j] + a[i*3+2]*inv[2*3+j];

    // loc_new = loc1 + K @ (loc2 - loc1)
    float d0 = l2[0]-l1[0], d1 = l2[1]-l1[1], d2 = l2[2]-l1[2];
    locn[0] = l1[0] + K[0]*d0 + K[1]*d1 + K[2]*d2;
    locn[1] = l1[1] + K[3]*d0 + K[4]*d1 + K[5]*d2;
    locn[2] = l1[2] + K[6]*d0 + K[7]*d1 + K[8]*d2;

    // cov_new = cov1 + K @ cov1   (faithful to source '+')
#pragma unroll
    for (int i = 0; i < 3; ++i)
#pragma unroll
        for (int j = 0; j < 3; ++j)
            covn[i*3+j] = a[i*3+j]
                        + K[i*3+0]*a[0*3+j] + K[i*3+1]*a[1*3+j] + K[i*3+2]*a[2*3+j];
}

__global__ __launch_bounds__(TILE)
void GaussianMerge_34591666602829_kernel(const float* __restrict__ loc1,
                                         const float* __restrict__ cov1,
                                         const float* __restrict__ loc2,
                                         const float* __restrict__ cov2,
                                         float* __restrict__ loc_out,
                                         float* __restrict__ cov_out,
                                         int n, int fastOK) {
    __shared__ __align__(16) float s_c1[TILE*9];   // 9216 B
    __shared__ __align__(16) float s_c2[TILE*9];   // 9216 B
    __shared__ __align__(16) float s_l1[TILE*3];   // 3072 B
    __shared__ __align__(16) float s_l2[TILE*3];   // 3072 B

    const int t  = threadIdx.x;
    const int e0 = blockIdx.x * TILE;

    if (fastOK && (e0 + TILE <= n)) {
        // ---- async coalesced b128 global -> LDS staging (non-temporal) ----
        const uint32_t c1_lds = lds_off32(s_c1);
        const uint32_t c2_lds = lds_off32(s_c2);
        const uint32_t l1_lds = lds_off32(s_l1);
        const uint32_t l2_lds = lds_off32(s_l2);
        const uint64_t c1_g = (uint64_t)(uintptr_t)(cov1 + (size_t)e0 * 9);
        const uint64_t c2_g = (uint64_t)(uintptr_t)(cov2 + (size_t)e0 * 9);
        const uint64_t l1_g = (uint64_t)(uintptr_t)(loc1 + (size_t)e0 * 3);
        const uint64_t l2_g = (uint64_t)(uintptr_t)(loc2 + (size_t)e0 * 3);

#pragma unroll
        for (int c = t; c < (TILE*9*4)/16; c += TILE) {          // 576 chunks each
            async_ld_b128(c1_lds + (uint32_t)c*16u, (uint32_t)c*16u, c1_g);
            async_ld_b128(c2_lds + (uint32_t)c*16u, (uint32_t)c*16u, c2_g);
        }
#pragma unroll
        for (int c = t; c < (TILE*3*4)/16; c += TILE) {          // 192 chunks each
            async_ld_b128(l1_lds + (uint32_t)c*16u, (uint32_t)c*16u, l1_g);
            async_ld_b128(l2_lds + (uint32_t)c*16u, (uint32_t)c*16u, l2_g);
        }
        wait_async0();
        __syncthreads();

        // ---- per-thread gather (odd strides 9/3 -> bank-conflict-free) ----
        float a[9], b[9], l1v[3], l2v[3];
#pragma unroll
        for (int k = 0; k < 9; ++k) { a[k] = s_c1[t*9 + k]; b[k] = s_c2[t*9 + k]; }
#pragma unroll
        for (int k = 0; k < 3; ++k) { l1v[k] = s_l1[t*3 + k]; l2v[k] = s_l2[t*3 + k]; }

        float ln[3], cn[9];
        merge_one(a, b, l1v, l2v, ln, cn);

        // write results into own LDS region (no cross-thread hazard)
#pragma unroll
        for (int k = 0; k < 9; ++k) s_c1[t*9 + k] = cn[k];
#pragma unroll
        for (int k = 0; k < 3; ++k) s_l1[t*3 + k] = ln[k];
        __syncthreads();

        // ---- async coalesced b128 LDS -> global stores (non-temporal) ----
        // No trailing s_wait_asynccnt: S_ENDPGM performs an implicit wait-idle.
        const uint64_t lo_g = (uint64_t)(uintptr_t)(loc_out + (size_t)e0 * 3);
        const uint64_t co_g = (uint64_t)(uintptr_t)(cov_out + (size_t)e0 * 9);
#pragma unroll
        for (int c = t; c < (TILE*3*4)/16; c += TILE)
            async_st_b128((uint32_t)c*16u, l1_lds + (uint32_t)c*16u, lo_g);
#pragma unroll
        for (int c = t; c < (TILE*9*4)/16; c += TILE)
            async_st_b128((uint32_t)c*16u, c1_lds + (uint32_t)c*16u, co_g);
    } else {
        // ---- tail / unaligned fallback: direct per-thread path ----
        const int e = e0 + t;
        if (e < n) {
            float a[9], b[9], l1v[3], l2v[3];
#pragma unroll
            for (int k = 0; k < 9; ++k) { a[k] = cov1[(size_t)e*9 + k]; b[k] = cov2[(size_t)e*9 + k]; }
#pragma unroll
            for (int k = 0; k < 3; ++k) { l1v[k] = loc1[(size_t)e*3 + k]; l2v[k] = loc2[(size_t)e*3 + k]; }
            float ln[3], cn[9];
            merge_one(a, b, l1v, l2v, ln, cn);
#pragma unroll
            for (int k = 0; k < 3; ++k) loc_out[(size_t)e*3 + k] = ln[k];
#pragma unroll
            for (int k = 0; k < 9; ++k) cov_out[(size_t)e*9 + k] = cn[k];
        }
    }
}

extern "C" void kernel_launch(void* const* d_in, const int* in_sizes, int n_in,
                              void* d_out, int out_size, void* d_ws, size_t ws_size,
                              hipStream_t stream) {
    (void)n_in; (void)out_size; (void)d_ws; (void)ws_size;
    const float* loc1 = (const float*)d_in[0];
    const float* cov1 = (const float*)d_in[1];
    const float* loc2 = (const float*)d_in[2];
    const float* cov2 = (const float*)d_in[3];

    const int n = in_sizes[0] / 3;            // B elements
    float* out     = (float*)d_out;
    float* loc_out = out;                      // [B,3]
    float* cov_out = out + (size_t)3 * n;      // [B,3,3]

    // cov_out base must be 16B-aligned for b128 async stores: 12*n % 16 == 0
    const int fastOK = ((n & 3) == 0) ? 1 : 0;

    const int nblocks = (n + TILE - 1) / TILE;
    GaussianMerge_34591666602829_kernel<<<nblocks, TILE, 0, stream>>>(
        loc1, cov1, loc2, cov2, loc_out, cov_out, n, fastOK);
}